// BSplineNN_56727928046496
// MI455X (gfx1250) — compile-verified
//
#include <hip/hip_runtime.h>
#include <cstdint>

#define BATCH  2048
#define FEAT   512
#define NB     64          // N_BSPLINES
#define NK     68          // knots per feature
#define WIN    5           // DEGREE + 2
#define NI     68          // interval slots (i = 0..67; 67 handles x >= last knot)
#define NMOM   4           // cubic: 4 Horner coefficients
#define GCOLS  (NI * NMOM) // 272
#define FT     64          // features per block (main kernel)
#define BT     32          // batch rows per block (main kernel)

typedef __attribute__((ext_vector_type(2))) float v2f;
typedef __attribute__((ext_vector_type(8))) float v8f;

// ---------------------------------------------------------------------------
// Kernel 1: build G (NB x GCOLS) from the (shared) knot vector.
// G[s, 4*i+m] = sum_j coef[s,j] * g_m(k_{s+j}) * [s+j <= i]
// with g_0=-k^3, g_1=3k^2, g_2=-3k, g_3=1  (Horner coeffs of relu(x-k)^3).
// ---------------------------------------------------------------------------
__global__ __launch_bounds__(256) void build_g_kernel(const float* __restrict__ knots,
                                                      float* __restrict__ G) {
  __shared__ float sk[NK];
  __shared__ float scoef[NB * WIN];
  const int tid = threadIdx.x;
  if (tid < NK) sk[tid] = knots[tid];  // feature-0 row (knots broadcast per reference)
  __syncthreads();
  for (int e = tid; e < NB * WIN; e += 256) {
    const int s = e / WIN, j = e % WIN;
    const float kj = sk[s + j];
    float p = 1.0f;
#pragma unroll
    for (int l = 0; l < WIN; ++l)
      if (l != j) p *= (sk[s + l] - kj);
    scoef[e] = 1.0f / p;
  }
  __syncthreads();
  for (int e = tid; e < NB * GCOLS; e += 256) {
    const int s = e / GCOLS;
    const int c = e % GCOLS;
    const int i = c >> 2;
    const int m = c & 3;
    float acc = 0.0f;
#pragma unroll
    for (int j = 0; j < WIN; ++j) {
      const int t = s + j;
      if (t <= i) {
        const float k = sk[t];
        float g;
        if      (m == 0) g = -k * k * k;
        else if (m == 1) g = 3.0f * k * k;
        else if (m == 2) g = -3.0f * k;
        else             g = 1.0f;
        acc = __builtin_fmaf(scoef[s * WIN + j], g, acc);
      }
    }
    G[e] = acc;
  }
}

// ---------------------------------------------------------------------------
// Kernel 2: P (FEAT x GCOLS) = W (FEAT x NB) @ G (NB x GCOLS)
// One wave32 per 16x16 output tile, V_WMMA_F32_16X16X4_F32, K = 64 in 16 steps.
// ---------------------------------------------------------------------------
__global__ __launch_bounds__(32) void table_gemm_kernel(const float* __restrict__ Wt,
                                                        const float* __restrict__ G,
                                                        float* __restrict__ P) {
  const int lane = threadIdx.x;
  const int row  = lane & 15;
  const int hi   = lane >> 4;
  const int tile = blockIdx.x;                  // (FEAT/16)*(GCOLS/16) tiles
  const int m0   = (tile / (GCOLS / 16)) * 16;
  const int n0   = (tile % (GCOLS / 16)) * 16;
  v8f c = {};
#pragma unroll
  for (int k0 = 0; k0 < NB; k0 += 4) {
    const int ka = k0 + 2 * hi;
    v2f a, b;
    a.x = Wt[(m0 + row) * NB + ka];
    a.y = Wt[(m0 + row) * NB + ka + 1];
    b.x = G[ka * GCOLS + n0 + row];
    b.y = G[(ka + 1) * GCOLS + n0 + row];
    c = __builtin_amdgcn_wmma_f32_16x16x4_f32(false, a, false, b, (short)0, c,
                                              false, false);
  }
#pragma unroll
  for (int r = 0; r < 8; ++r)
    P[(size_t)(m0 + r + 8 * hi) * GCOLS + n0 + row] = c[r];
}

// ---------------------------------------------------------------------------
// Kernel 3: out[b,f] = Horner(P[f, interval(x)], x).
// Stages the block's 64-feature table slice (69.6 KB, contiguous) into LDS via
// global_load_async_to_lds_b128 (ASYNCcnt), then does the data-dependent
// per-interval gather from LDS.
// ---------------------------------------------------------------------------
__global__ __launch_bounds__(256) void bspline_eval_kernel(const float* __restrict__ x,
                                                           const float* __restrict__ knots,
                                                           const float* __restrict__ P,
                                                           float* __restrict__ out) {
  extern __shared__ unsigned char dynsmem[];
  float4* sP    = (float4*)dynsmem;                       // FT*NI float4 = 69632 B
  float2* sMeta = (float2*)(dynsmem + (size_t)FT * NI * sizeof(float4));

  const int tid = threadIdx.x;
  const int f0  = blockIdx.x * FT;
  const int b0  = blockIdx.y * BT;

  // Async copy of the contiguous 64-row slice of P into LDS (bypasses VGPRs).
  const float4* src = (const float4*)P + (size_t)f0 * NI;
#pragma unroll
  for (int it = 0; it < (FT * NI) / 256; ++it) {
    const int idx = it * 256 + tid;
    const uint32_t ldsaddr = (uint32_t)(uintptr_t)(sP + idx);
    const unsigned long long gaddr = (unsigned long long)(uintptr_t)(src + idx);
    asm volatile("global_load_async_to_lds_b128 %0, %1, off"
                 :: "v"(ldsaddr), "v"(gaddr) : "memory");
  }
  if (tid < FT) {  // per-feature uniform-grid parameters from its own knot row
    const float k0 = knots[(size_t)(f0 + tid) * NK + 0];
    const float k1 = knots[(size_t)(f0 + tid) * NK + 1];
    sMeta[tid] = make_float2(k0, 1.0f / (k1 - k0));
  }
  asm volatile("s_wait_asynccnt 0" ::: "memory");
  __syncthreads();

  const int lf = tid & (FT - 1);   // feature lane -> coalesced 128B rows
  const int rb = tid >> 6;         // 0..3
  const float2 meta = sMeta[lf];
  const float4* rowP = sP + (size_t)lf * NI;

#pragma unroll
  for (int it = 0; it < BT / 4; ++it) {
    const int b = b0 + rb + it * 4;
    const size_t off = (size_t)b * FEAT + f0 + lf;
    const float xv = x[off];
    int i = (int)((xv - meta.x) * meta.y);
    i = i < 0 ? 0 : (i > NI - 1 ? NI - 1 : i);
    const float4 A = rowP[i];                        // ds_load_b128 gather
    float r = __builtin_fmaf(xv, A.w, A.z);
    r = __builtin_fmaf(xv, r, A.y);
    r = __builtin_fmaf(xv, r, A.x);
    out[off] = r;
  }
}

extern "C" void kernel_launch(void* const* d_in, const int* in_sizes, int n_in,
                              void* d_out, int out_size, void* d_ws, size_t ws_size,
                              hipStream_t stream) {
  (void)in_sizes; (void)n_in; (void)out_size; (void)ws_size;
  const float* x     = (const float*)d_in[0];   // (2048, 512)
  const float* knots = (const float*)d_in[1];   // (512, 68)
  const float* w     = (const float*)d_in[2];   // (512, 64)
  float* out = (float*)d_out;                   // (2048, 512)

  float* G = (float*)d_ws;                      // 64 x 272
  float* P = G + (size_t)NB * GCOLS;            // 512 x 272 (16B-aligned offset)

  build_g_kernel<<<1, 256, 0, stream>>>(knots, G);
  table_gemm_kernel<<<(FEAT / 16) * (GCOLS / 16), 32, 0, stream>>>(w, G, P);

  const size_t smem = (size_t)FT * NI * sizeof(float4) + FT * sizeof(float2);
  bspline_eval_kernel<<<dim3(FEAT / FT, BATCH / BT), 256, smem, stream>>>(x, knots, P, out);
}